// OpenMamba_30064771072282
// MI455X (gfx1250) — compile-verified
//
#include <hip/hip_runtime.h>
#include <math.h>

#define D_MODEL   768
#define D_STATE   64
#define HEADDIM   64
#define D_INNER   1536
#define NHEADS    24
#define CONV_CH   1664
#define D_IN_PROJ 3224          // 2*1536 + 2*64 + 24
#define SEQ_LEN   256
#define BATCH     8
#define ROWS      (SEQ_LEN*BATCH)   // 2048
#define WS_LDZ    3232              // padded row stride (101*32)
#define EPS       1e-5f

typedef __attribute__((ext_vector_type(2))) float v2f;
typedef __attribute__((ext_vector_type(8))) float v8f;

// Explicit global-address-space pointers so GEMM loads/stores lower to
// global_load_*/global_store_* (not flat_*, which drags DScnt into every wait).
typedef const float __attribute__((address_space(1)))* gcfp;
typedef float       __attribute__((address_space(1)))* gfp;

__device__ __forceinline__ float silu_f(float x) {
    return x / (1.f + __expf(-x));
}

// ---------------------------------------------------------------------------
// Register-blocked fp32 WMMA GEMM: each wave computes a 64(M) x 32(N) macro-
// tile as 4x2 grid of 16x16 WMMA tiles. Per K-step of 4:
//   4 x global_load_b64 (A fragments) + 4 x global_load_b32 (B fragments)
//   -> 8 x v_wmma_f32_16x16x4_f32   (1 VMEM op per WMMA)
// A fragment reused across the 2 N-tiles; B fragment across the 4 M-tiles.
// Lane layouts per CDNA5 ISA:
//   A (16x4):  lane<16 -> M=lane, K=k+0/k+1 ; lane>=16 -> K=k+2/k+3
//   B (4x16):  VGPR j  -> K=k+2*(lane>>4)+j, N=lane&15
//   C/D:       VGPR r  -> M=r+8*(lane>>4),   N=lane&15
// Pad columns (>= N) are clamped on load and masked on store.
// ---------------------------------------------------------------------------
__global__ void wmma_gemm_f32(const float* __restrict__ A,
                              const float* __restrict__ B,
                              float* __restrict__ C,
                              int N, int K,
                              int lda, int ldb, int ldc,
                              int ngrp, int waves_total)
{
    const int wave = threadIdx.x >> 5;
    const int lane = threadIdx.x & 31;
    const int wid  = blockIdx.x * (blockDim.x >> 5) + wave;
    if (wid >= waves_total) return;            // wave-uniform exit

    const int tm = wid / ngrp;                 // 64-row block index
    const int tn = wid - tm * ngrp;            // 32-col block index

    const int l15   = lane & 15;
    const int khalf = (lane >> 4) * 2;         // 0 or 2

    gcfp Ag = (gcfp)A;
    gcfp Bg = (gcfp)B;

    size_t aoff[4];
    #pragma unroll
    for (int mi = 0; mi < 4; ++mi)
        aoff[mi] = (size_t)(tm * 64 + mi * 16 + l15) * lda + khalf;

    size_t boff[2];
    #pragma unroll
    for (int ni = 0; ni < 2; ++ni) {
        int c = tn * 32 + ni * 16 + l15;
        c = c < N ? c : N - 1;                 // clamp pad columns
        boff[ni] = (size_t)khalf * ldb + c;
    }

    v8f acc[4][2];
    #pragma unroll
    for (int mi = 0; mi < 4; ++mi)
        #pragma unroll
        for (int ni = 0; ni < 2; ++ni)
            acc[mi][ni] = {};

    const size_t bstep = (size_t)4 * ldb;

    for (int k = 0; k < K; k += 4) {
        v2f a[4], b[2];
        #pragma unroll
        for (int mi = 0; mi < 4; ++mi) {       // contiguous pair -> b64 load
            a[mi][0] = Ag[aoff[mi]];
            a[mi][1] = Ag[aoff[mi] + 1];
        }
        #pragma unroll
        for (int ni = 0; ni < 2; ++ni) {
            b[ni][0] = Bg[boff[ni]];
            b[ni][1] = Bg[boff[ni] + ldb];
        }
        #pragma unroll
        for (int mi = 0; mi < 4; ++mi)
            #pragma unroll
            for (int ni = 0; ni < 2; ++ni)
                acc[mi][ni] = __builtin_amdgcn_wmma_f32_16x16x4_f32(
                    false, a[mi], false, b[ni],
                    (short)0, acc[mi][ni], false, false);
        #pragma unroll
        for (int mi = 0; mi < 4; ++mi) aoff[mi] += 4;
        #pragma unroll
        for (int ni = 0; ni < 2; ++ni) boff[ni] += bstep;
    }

    gfp Cg = (gfp)C;
    #pragma unroll
    for (int ni = 0; ni < 2; ++ni) {
        const int ccol = tn * 32 + ni * 16 + l15;
        if (ccol < N) {
            #pragma unroll
            for (int mi = 0; mi < 4; ++mi) {
                const int crow0 = tm * 64 + mi * 16 + (lane >> 4) * 8;
                #pragma unroll
                for (int r = 0; r < 8; ++r)
                    Cg[(size_t)(crow0 + r) * ldc + ccol] = acc[mi][ni][r];
            }
        }
    }
}

// ---------------------------------------------------------------------------
// Sequential scan. One block per (batch, head) pair (192 blocks, 256 thr).
// State h[64,64] lives in registers: thread (p = tid>>2, q = tid&3) owns
// h[p][q*16 .. q*16+15]. Per step: stage conv'd x/B/C rows + dt scalars in
// LDS, update state, dump state to the states output (805 MB total -> the
// HBM-bound part), reduce y over 4 lanes, gate with silu(z), write ybuf.
// ---------------------------------------------------------------------------
__global__ void mamba_scan(const float* __restrict__ zx,      // [ROWS, WS_LDZ]
                           const float* __restrict__ conv_w,  // [CONV_CH,4]
                           const float* __restrict__ conv_b,  // [CONV_CH]
                           const float* __restrict__ dt_bias, // [NHEADS]
                           const float* __restrict__ A_log,   // [NHEADS]
                           const float* __restrict__ Dp,      // [NHEADS]
                           float* __restrict__ ybuf,          // [ROWS, D_INNER]
                           float* __restrict__ states)        // [SEQ,B,NH,64,64]
{
    const int b    = blockIdx.x & 7;
    const int head = blockIdx.x >> 3;
    const int tid  = threadIdx.x;
    const int p    = tid >> 2;
    const int q    = tid & 3;
    const int n0   = q * 16;

    __shared__ float sX[64], sB[64], sC[64];
    __shared__ float sDT, sDA;

    const float Ahead = -__expf(A_log[head]);
    const float dhead = Dp[head];
    const float dtb   = dt_bias[head];

    float h[16];
    #pragma unroll
    for (int j = 0; j < 16; ++j) h[j] = 0.f;

    gcfp zxg = (gcfp)zx;
    gfp  stg = (gfp)states;
    gfp  ybg = (gfp)ybuf;

    for (int t = 0; t < SEQ_LEN; ++t) {
        const size_t rbase = (size_t)(t * BATCH + b) * WS_LDZ;

        if (tid < 64) {                         // x slice for this head
            const int c = head * 64 + tid;      // conv channel
            float v = zxg[rbase + D_INNER + c];
            v = v * conv_w[c * 4 + 3] + conv_b[c];
            sX[tid] = silu_f(v);
        } else if (tid < 128) {                 // B (d_state)
            const int n = tid - 64;
            const int c = D_INNER + n;          // conv channel 1536+n
            float v = zxg[rbase + D_INNER + c];
            v = v * conv_w[c * 4 + 3] + conv_b[c];
            sB[n] = silu_f(v);
        } else if (tid < 192) {                 // C (d_state)
            const int n = tid - 128;
            const int c = D_INNER + D_STATE + n;
            float v = zxg[rbase + D_INNER + c];
            v = v * conv_w[c * 4 + 3] + conv_b[c];
            sC[n] = silu_f(v);
        } else if (tid == 192) {                // dt scalars
            float dtv = zxg[rbase + D_INNER + CONV_CH + head] + dtb;
            dtv = dtv > 20.f ? dtv : log1pf(__expf(dtv));
            sDT = dtv;
            sDA = __expf(dtv * Ahead);
        }
        __syncthreads();

        const float dA   = sDA;
        const float xv   = sX[p];
        const float coef = sDT * xv;

        float y = 0.f;
        #pragma unroll
        for (int j = 0; j < 16; ++j) {
            h[j] = h[j] * dA + coef * sB[n0 + j];
            y += h[j] * sC[n0 + j];
        }
        y += __shfl_xor(y, 1, 32);
        y += __shfl_xor(y, 2, 32);

        // stream updated state to output (dominant HBM traffic)
        const size_t hbase = (((size_t)(t * BATCH + b) * NHEADS + head) << 12)
                             + p * 64 + n0;
        #pragma unroll
        for (int j = 0; j < 16; ++j)
            stg[hbase + j] = h[j];

        if (q == 0) {
            const int c = head * 64 + p;
            const float zraw = zxg[rbase + c];
            const float yy = (y + dhead * xv) * silu_f(zraw);
            ybg[(size_t)(t * BATCH + b) * D_INNER + c] = yy;
        }
        __syncthreads();
    }
}

// ---------------------------------------------------------------------------
// RMS-norm each row of ybuf in place (2048 rows x 1536), apply norm_w.
// ---------------------------------------------------------------------------
__global__ void rmsnorm_rows(float* __restrict__ y, const float* __restrict__ w)
{
    const int r   = blockIdx.x;
    float* row    = y + (size_t)r * D_INNER;
    const int tid = threadIdx.x;

    float v[6];
    float ss = 0.f;
    #pragma unroll
    for (int j = 0; j < 6; ++j) {
        v[j] = row[tid + j * 256];
        ss += v[j] * v[j];
    }
    #pragma unroll
    for (int o = 16; o > 0; o >>= 1) ss += __shfl_xor(ss, o, 32);

    __shared__ float sW[8];
    if ((tid & 31) == 0) sW[tid >> 5] = ss;
    __syncthreads();

    float tot = 0.f;
    #pragma unroll
    for (int i = 0; i < 8; ++i) tot += sW[i];

    const float scale = rsqrtf(tot * (1.f / D_INNER) + EPS);
    #pragma unroll
    for (int j = 0; j < 6; ++j) {
        const int c = tid + j * 256;
        row[c] = v[j] * scale * w[c];
    }
}

// ---------------------------------------------------------------------------
extern "C" void kernel_launch(void* const* d_in, const int* in_sizes, int n_in,
                              void* d_out, int out_size, void* d_ws, size_t ws_size,
                              hipStream_t stream) {
    const float* seq     = (const float*)d_in[0];
    const float* W_in    = (const float*)d_in[1];
    const float* conv_w  = (const float*)d_in[2];
    const float* conv_b  = (const float*)d_in[3];
    const float* dt_bias = (const float*)d_in[4];
    const float* A_log   = (const float*)d_in[5];
    const float* Dv      = (const float*)d_in[6];
    const float* norm_w  = (const float*)d_in[7];
    const float* W_out   = (const float*)d_in[8];

    float* out    = (float*)d_out;                                   // [256,8,768]
    float* states = out + (size_t)SEQ_LEN * BATCH * D_MODEL;         // [256,8,24,64,64]

    float* ws_z = (float*)d_ws;                                      // [2048, 3232]
    float* ws_y = ws_z + (size_t)ROWS * WS_LDZ;                      // [2048, 1536]

    // Phase A: in-projection GEMM [2048x768]@[768x3224] -> ws_z
    {
        const int ngrp        = (D_IN_PROJ + 31) / 32;               // 101
        const int waves_total = (ROWS / 64) * ngrp;                  // 3232
        const int blocks      = (waves_total + 7) / 8;
        wmma_gemm_f32<<<blocks, 256, 0, stream>>>(
            seq, W_in, ws_z, D_IN_PROJ, D_MODEL,
            D_MODEL, D_IN_PROJ, WS_LDZ, ngrp, waves_total);
    }

    // Phase B: sequential scan over t, parallel over (batch, head)
    mamba_scan<<<NHEADS * BATCH, 256, 0, stream>>>(
        ws_z, conv_w, conv_b, dt_bias, A_log, Dv, ws_y, states);

    // Phase C1: RMS norm rows in place
    rmsnorm_rows<<<ROWS, 256, 0, stream>>>(ws_y, norm_w);

    // Phase C2: out-projection GEMM [2048x1536]@[1536x768] -> outputs
    {
        const int ngrp        = D_MODEL / 32;                        // 24
        const int waves_total = (ROWS / 64) * ngrp;                  // 768
        const int blocks      = (waves_total + 7) / 8;
        wmma_gemm_f32<<<blocks, 256, 0, stream>>>(
            ws_y, W_out, out, D_MODEL, D_INNER,
            D_INNER, D_MODEL, D_MODEL, ngrp, waves_total);
    }
}